// LiftSplatShoot_33990371181068
// MI455X (gfx1250) — compile-verified
//
#include <hip/hip_runtime.h>
#include <math.h>

// ---------------------------------------------------------------------------
// Lift-Splat-Shoot voxel pooling for MI455X (gfx1250, wave32)
//   B=4 N=6 D=59 C=64 FH=16 FW=44  -> out (4, 64, 200, 200) f32
// ---------------------------------------------------------------------------

typedef __attribute__((ext_vector_type(2))) float v2f;
typedef __attribute__((ext_vector_type(8))) float v8f;

#define D_BINS   59
#define CAMC     64
#define B_       4
#define N_       6
#define FH_      16
#define FW_      44
#define CH_      (D_BINS + CAMC)       // 123
#define SPATIAL  (FH_ * FW_)           // 704
#define NPIX     (B_ * N_ * SPATIAL)   // 16896
#define OUT_HW   (200 * 200)           // 40000
#define OUT_BST  (CAMC * OUT_HW)       // 2560000

__device__ __forceinline__ float readlane_f(float v, int lane) {
  return __int_as_float(__builtin_amdgcn_readlane(__float_as_int(v), lane));
}

// --------------------------------------------------------------------------
// Kernel 1: per-(b,n) matrix   M = [ rot @ inv(K) | trans ]   (3x4 row-major,
// padded to 4 floats/row -> 12 floats per camera) into workspace.
// --------------------------------------------------------------------------
__global__ void lss_setup(const float* __restrict__ intr,
                          const float* __restrict__ extr,
                          float* __restrict__ mats) {
  int t = threadIdx.x;
  if (t >= B_ * N_) return;
  const float* K = intr + t * 9;
  const float* E = extr + t * 16;
  float a = K[0], b = K[1], c = K[2];
  float d = K[3], e = K[4], f = K[5];
  float g = K[6], h = K[7], i = K[8];
  float A0 = e*i - f*h, A1 = c*h - b*i, A2 = b*f - c*e;
  float A3 = f*g - d*i, A4 = a*i - c*g, A5 = c*d - a*f;
  float A6 = d*h - e*g, A7 = b*g - a*h, A8 = a*e - b*d;
  float invdet = 1.0f / (a*A0 + b*A3 + c*A6);
  float iK[9] = {A0*invdet, A1*invdet, A2*invdet,
                 A3*invdet, A4*invdet, A5*invdet,
                 A6*invdet, A7*invdet, A8*invdet};
  for (int r = 0; r < 3; ++r) {
    for (int cc = 0; cc < 3; ++cc) {
      float s = 0.0f;
      for (int k = 0; k < 3; ++k) s += E[r*4 + k] * iK[k*3 + cc];
      mats[t*12 + r*4 + cc] = s;
    }
    mats[t*12 + r*4 + 3] = E[r*4 + 3];   // translation
  }
}

// --------------------------------------------------------------------------
// Kernel 2: zero the 41 MB accumulator (float4 stores).
// --------------------------------------------------------------------------
__global__ void lss_zero(float4* __restrict__ out, int n4) {
  int i = blockIdx.x * blockDim.x + threadIdx.x;
  if (i < n4) out[i] = make_float4(0.f, 0.f, 0.f, 0.f);
}

// --------------------------------------------------------------------------
// Kernel 3: fused geometry (WMMA f32 16x16x4) + softmax + atomic scatter.
// One wave32 per pixel (b,n,h,w); 8 waves / block; grid sized exactly so
// every wave is fully active (EXEC all-1s, required for WMMA).
// --------------------------------------------------------------------------
__global__ void __launch_bounds__(256)
lss_main(const float* __restrict__ cam,
         const float* __restrict__ frustum,
         const float* __restrict__ mats,
         float* __restrict__ out) {
  const int lane = threadIdx.x & 31;
  const int wave = blockIdx.x * (blockDim.x >> 5) + (threadIdx.x >> 5);  // pixel id
  const int w  = wave % FW_;
  const int h  = (wave / FW_) % FH_;
  const int bn = wave / SPATIAL;
  const int b  = bn / N_;

  // ---- A-matrix (16x4): rows 0..2 = [comb|trans], rows 3..15 = 0.
  // ISA layout: VGPR0 = K=0 (lanes 0-15) / K=2 (lanes 16-31); VGPR1 = K=1 / K=3.
  const float* M = mats + bn * 12;
  const int row   = lane & 15;
  const int khalf = (lane >> 4) << 1;                 // 0 or 2
  v2f Am;
  Am.x = (row < 3) ? M[row*4 + khalf]     : 0.0f;
  Am.y = (row < 3) ? M[row*4 + khalf + 1] : 0.0f;

  // ---- geometry for 64 depth slots via 4 WMMAs; per tile t, lane j<16 gets
  // the flattened output offset for depth d = 16*t + j (or -1 if masked).
  int offs[4];
#pragma unroll
  for (int t = 0; t < 4; ++t) {
    const int j  = lane & 15;
    const int dd = t * 16 + j;
    const int dc = (dd < D_BINS) ? dd : (D_BINS - 1);       // clamp pad lanes
    const float* F = frustum + ((size_t)(dc * FH_ + h) * FW_ + w) * 3;
    const float fx = F[0], fy = F[1], fz = F[2];
    // B-matrix (4x16): lanes 0-15 -> K0 = x*d, K1 = y*d ; lanes 16-31 -> K2 = d, K3 = 1
    v2f Bm;
    Bm.x = (lane < 16) ? fx * fz : fz;
    Bm.y = (lane < 16) ? fy * fz : 1.0f;
    v8f Cm = {};
    v8f Dm = __builtin_amdgcn_wmma_f32_16x16x4_f32(
        /*neg_a=*/false, Am, /*neg_b=*/false, Bm,
        /*c_mod=*/(short)0, Cm, /*reuse_a=*/false, /*reuse_b=*/false);
    // lanes 0-15: Dm[0..2] = geom x,y,z for point j
    const float gx = Dm[0], gy = Dm[1], gz = Dm[2];
    // trunc-toward-zero voxelization (matches jnp astype(int32)); /0.5 is an
    // exact *2.  vz==0  <=>  trunc((gz+10)/20)==0  <=>  |gz+10| < 20.
    const int vx = (int)((gx + 50.0f) / 0.5f);
    const int vy = (int)((gy + 50.0f) / 0.5f);
    const bool validz = fabsf(gz + 10.0f) < 20.0f;
    const bool valid = (dd < D_BINS) && validz &&
                       (vx >= 0) && (vx < 200) &&
                       (vy >= 0) && (vy < 200);
    // output is flipped: out[b, c, 199-vx, 199-vy]
    offs[t] = valid ? (b * OUT_BST + (199 - vx) * 200 + (199 - vy)) : -1;
  }

  // ---- softmax over the 59 depth logits (2 per lane, shfl reductions)
  const float* base = cam + (size_t)bn * CH_ * SPATIAL + h * FW_ + w;
  const float NEG_INF = -__builtin_inff();
  float l0 = base[lane * SPATIAL];
  float l1 = (lane + 32 < D_BINS) ? base[(lane + 32) * SPATIAL] : NEG_INF;
  float m = fmaxf(l0, l1);
#pragma unroll
  for (int o = 16; o; o >>= 1) m = fmaxf(m, __shfl_xor(m, o, 32));
  float e0 = __expf(l0 - m);
  float e1 = (lane + 32 < D_BINS) ? __expf(l1 - m) : 0.0f;
  float s = e0 + e1;
#pragma unroll
  for (int o = 16; o; o >>= 1) s += __shfl_xor(s, o, 32);
  const float invs = 1.0f / s;

  // ---- features (c = lane, lane+32), pre-scaled by 1/sum
  const float f0 = base[(D_BINS + lane)      * SPATIAL] * invs;
  const float f1 = base[(D_BINS + 32 + lane) * SPATIAL] * invs;

  // ---- scatter: 59 depth bins x 64 channels of L2-resident f32 atomics.
  // Broadcasts use v_readlane (immediate lane) -> off is wave-uniform, the
  // guard becomes a scalar branch, and per-lane channel bases are hoisted.
  float* const outc0 = out + lane * OUT_HW;
  float* const outc1 = out + (lane + 32) * OUT_HW;
#pragma unroll
  for (int t = 0; t < 4; ++t) {
    const int dmax = (t == 3) ? (D_BINS - 48) : 16;   // 16,16,16,11
#pragma unroll
    for (int j = 0; j < dmax; ++j) {
      const int d = t * 16 + j;
      const float ed = (d < 32) ? readlane_f(e0, d) : readlane_f(e1, d - 32);
      const int off = __builtin_amdgcn_readlane(offs[t], j);
      if (off >= 0) {
        unsafeAtomicAdd(outc0 + off, ed * f0);
        unsafeAtomicAdd(outc1 + off, ed * f1);
      }
    }
  }
}

// --------------------------------------------------------------------------
extern "C" void kernel_launch(void* const* d_in, const int* in_sizes, int n_in,
                              void* d_out, int out_size, void* d_ws, size_t ws_size,
                              hipStream_t stream) {
  (void)in_sizes; (void)n_in; (void)ws_size;
  const float* cam     = (const float*)d_in[0];   // (B,N,123,16,44)
  const float* intr    = (const float*)d_in[1];   // (B,N,3,3)
  const float* extr    = (const float*)d_in[2];   // (B,N,4,4)
  const float* frustum = (const float*)d_in[3];   // (59,16,44,3)
  float* out  = (float*)d_out;                    // (4,64,200,200)
  float* mats = (float*)d_ws;                     // 24 * 12 floats

  lss_setup<<<1, 32, 0, stream>>>(intr, extr, mats);

  const int n4 = out_size / 4;                    // 2,560,000 float4
  lss_zero<<<(n4 + 255) / 256, 256, 0, stream>>>((float4*)out, n4);

  lss_main<<<NPIX / 8, 256, 0, stream>>>(cam, frustum, mats, out);
}